// GNN_28269474743135
// MI455X (gfx1250) — compile-verified
//
#include <hip/hip_runtime.h>

typedef __attribute__((ext_vector_type(2))) float v2f;
typedef __attribute__((ext_vector_type(8))) float v8f;

#define NEG_SLOPE 0.2f

// ---- ordered-uint mapping for float atomic max (memset-0 == -infinity sentinel) ----
__device__ __forceinline__ unsigned fmap(float f) {
  unsigned u = __float_as_uint(f);
  return (u & 0x80000000u) ? ~u : (u | 0x80000000u);
}
__device__ __forceinline__ float funmap(unsigned u) {
  unsigned v = (u & 0x80000000u) ? (u & 0x7fffffffu) : ~u;
  return __uint_as_float(v);
}

// =====================================================================
// WMMA f32 GEMM: C[M,NC] = A[M,K] * B[K,NC]; one wave per 16x16 tile.
// Uses V_WMMA_F32_16X16X4_F32 (CDNA5, wave32, full fp32 precision).
// K and NC are compile-time: store offsets r*NC become immediates and
// B addressing uses shifts, so the epilogue is 8 clean global_store_b32.
// A frag (16x4): lane m=lane&15, K pair = k0+2*(lane>>4) -> contiguous float2.
// B frag (4x16): lanes0-15 hold K={k0,k0+1}, lanes16-31 hold K={k0+2,k0+3}.
// C/D (16x16): VGPR r -> row = r + 8*(lane>>4), col = lane&15.
// =====================================================================
template<int K, int NC>
__global__ void gemm_wmma_f32(const float* __restrict__ A, const float* __restrict__ B,
                              float* __restrict__ C, int M) {
  int wave = (blockIdx.x * blockDim.x + threadIdx.x) >> 5;
  int lane = threadIdx.x & 31;
  constexpr int nct = NC >> 4;
  int ntiles = ((M + 15) >> 4) * nct;
  if (wave >= ntiles) return;                 // wave-uniform exit: EXEC all-1s inside
  int tm = wave / nct, tc = wave - tm * nct;
  int row_base = tm << 4, col_base = tc << 4;
  int mrow = lane & 15, hi = lane >> 4;

  int arow = row_base + mrow; if (arow > M - 1) arow = M - 1;   // clamp (M%16==0 here)
  const float* Arow = A + (size_t)arow * K;
  const float* Bcol = B + col_base + mrow;

  v8f acc = {};
#pragma unroll 8
  for (int k0 = 0; k0 < K; k0 += 4) {
    int ka = k0 + 2 * hi;
    v2f a, b;
    a.x = Arow[ka];
    a.y = Arow[ka + 1];
    b.x = Bcol[ka * NC];
    b.y = Bcol[(ka + 1) * NC];
    acc = __builtin_amdgcn_wmma_f32_16x16x4_f32(
        /*neg_a=*/false, a, /*neg_b=*/false, b,
        /*c_mod=*/(short)0, acc, /*reuse_a=*/false, /*reuse_b=*/false);
  }

  // Epilogue: base pointer once, 8 stores at compile-time offsets r*NC.
  int rbase = row_base + hi * 8;
  float* p = C + (size_t)rbase * NC + col_base + mrow;
  if (rbase + 8 <= M) {
#pragma unroll
    for (int r = 0; r < 8; ++r) p[r * NC] = acc[r];
  } else {
#pragma unroll
    for (int r = 0; r < 8; ++r) { if (rbase + r < M) p[r * NC] = acc[r]; }
  }
}

// =====================================================================
// alpha_src/alpha_dst per (node, head): dot(xW[n,h,:64], a_vec[h,:64]).
// One wave per (n,h): lane covers c=lane and c=lane+32, then shfl-xor reduce.
// =====================================================================
__global__ void node_alpha(const float* __restrict__ xW,
                           const float* __restrict__ avs, const float* __restrict__ avd,
                           float* __restrict__ as, float* __restrict__ ad, int N, int H) {
  int wave = (blockIdx.x * blockDim.x + threadIdx.x) >> 5;
  int lane = threadIdx.x & 31;
  if (wave >= N * H) return;
  int h = wave % H;
  const float* row = xW + (size_t)wave * 64;
  const float* vs = avs + h * 64;
  const float* vd = avd + h * 64;
  float s = row[lane] * vs[lane] + row[lane + 32] * vs[lane + 32];
  float d = row[lane] * vd[lane] + row[lane + 32] * vd[lane + 32];
#pragma unroll
  for (int off = 16; off; off >>= 1) {
    s += __shfl_xor(s, off, 32);
    d += __shfl_xor(d, off, 32);
  }
  if (lane == 0) { as[wave] = s; ad[wave] = d; }
}

// =====================================================================
// Per (edge, head): e = leaky_relu(as[src]+ad[dst]); store e; atomicMax into m.
// =====================================================================
template<int H>
__global__ void edge_leaky_max(const float* __restrict__ as, const float* __restrict__ ad,
                               const int* __restrict__ ei, float* __restrict__ ebuf,
                               unsigned* __restrict__ mU, int E) {
  int gid = blockIdx.x * blockDim.x + threadIdx.x;
  if (gid >= E * H) return;
  int e = gid / H, h = gid - e * H;
  int s = ei[e], d = ei[E + e];
  float v = as[s * H + h] + ad[d * H + h];
  v = (v >= 0.f) ? v : NEG_SLOPE * v;
  ebuf[gid] = v;
  atomicMax(&mU[d * H + h], fmap(v));
}

__global__ void decode_max(unsigned* __restrict__ mU, int n) {
  int gid = blockIdx.x * blockDim.x + threadIdx.x;
  if (gid >= n) return;
  float f = funmap(mU[gid]);
  ((float*)mU)[gid] = f;  // in-place: buffer becomes float m
}

// Per (edge, head): ex = exp(e - m[dst]); store; atomicAdd denom[dst].
template<int H>
__global__ void edge_expsum(float* __restrict__ ebuf, const float* __restrict__ mf,
                            const int* __restrict__ ei, float* __restrict__ denom, int E) {
  int gid = blockIdx.x * blockDim.x + threadIdx.x;
  if (gid >= E * H) return;
  int e = gid / H, h = gid - e * H;
  int d = ei[E + e];
  float ex = __expf(ebuf[gid] - mf[d * H + h]);
  ebuf[gid] = ex;
  atomicAdd(&denom[d * H + h], ex);
}

// Per (edge, head): alpha = ex / denom[dst]  (folded so hot kernel reads 1 scalar).
template<int H>
__global__ void edge_norm(float* __restrict__ ebuf, const float* __restrict__ denom,
                          const int* __restrict__ ei, int E) {
  int gid = blockIdx.x * blockDim.x + threadIdx.x;
  if (gid >= E * H) return;
  int e = gid / H, h = gid - e * H;
  int d = ei[E + e];
  ebuf[gid] = ebuf[gid] / denom[d * H + h];
}

// =====================================================================
// Heavy kernel: out[dst] += alpha * xW[src]. One thread per float4.
// xW (102 MB) is L2-resident on MI455X (192 MB L2) -> gathers/atomics hit L2.
// =====================================================================
template<int H, int C>
__global__ void edge_aggregate(const float* __restrict__ xW, const float* __restrict__ alpha,
                               const int* __restrict__ ei, float* __restrict__ out, int E) {
  constexpr int CH4 = (H * C) / 4;          // float4 chunks per edge
  int gid = blockIdx.x * blockDim.x + threadIdx.x;
  if (gid >= E * CH4) return;
  int e = gid / CH4, j4 = gid - e * CH4;
  int h = j4 / (C / 4);
  int s = ei[e], d = ei[E + e];
  float a = alpha[e * H + h];
  const float4* src4 = (const float4*)(xW + (size_t)s * (H * C));
  float4 v = src4[j4];
  float* o = out + (size_t)d * (H * C) + j4 * 4;
  atomicAdd(o + 0, v.x * a);
  atomicAdd(o + 1, v.y * a);
  atomicAdd(o + 2, v.z * a);
  atomicAdd(o + 3, v.w * a);
}

// bias add (+ optional ReLU), F is power of two
template<int F, bool RELU>
__global__ void bias_act(float* __restrict__ x, const float* __restrict__ b, int total) {
  int gid = blockIdx.x * blockDim.x + threadIdx.x;
  if (gid >= total) return;
  float v = x[gid] + b[gid & (F - 1)];
  if (RELU) v = v > 0.f ? v : 0.f;
  x[gid] = v;
}

// =====================================================================
extern "C" void kernel_launch(void* const* d_in, const int* in_sizes, int n_in,
                              void* d_out, int out_size, void* d_ws, size_t ws_size,
                              hipStream_t stream) {
  const float* x     = (const float*)d_in[0];
  const int*   ei    = (const int*)  d_in[1];
  const float* W1    = (const float*)d_in[2];
  const float* asrc1 = (const float*)d_in[3];
  const float* adst1 = (const float*)d_in[4];
  const float* b1    = (const float*)d_in[5];
  const float* W2    = (const float*)d_in[6];
  const float* asrc2 = (const float*)d_in[7];
  const float* adst2 = (const float*)d_in[8];
  const float* b2    = (const float*)d_in[9];
  float* out = (float*)d_out;

  const int D = 64;
  const int N = in_sizes[0] / D;           // 50000
  const int E = in_sizes[1] / 2;           // 500000
  const int H = 8;
  const int F1 = H * D;                    // 512

  // ---- workspace layout (floats) ----
  float* ws   = (float*)d_ws;
  float* xW1  = ws;                          // N*512
  float* out1 = xW1  + (size_t)N * F1;       // N*512 (becomes h after relu)
  float* e1   = out1 + (size_t)N * F1;       // E*8
  float* as1  = e1   + (size_t)E * H;        // N*8
  float* ad1  = as1  + (size_t)N * H;        // N*8
  float* m1   = ad1  + (size_t)N * H;        // N*8 (uint then float)
  float* dn1  = m1   + (size_t)N * H;        // N*8
  float* hW2  = dn1  + (size_t)N * H;        // N*64
  float* e2   = hW2  + (size_t)N * D;        // E
  float* as2  = e2   + (size_t)E;            // N
  float* ad2  = as2  + (size_t)N;            // N
  float* m2   = ad2  + (size_t)N;            // N (uint then float)
  float* dn2  = m2   + (size_t)N;            // N

  const int T = 256;

  // ---- zero-init accumulators / sentinels (memset 0 == -inf under fmap) ----
  hipMemsetAsync(out1, 0, (size_t)N * F1 * 4, stream);
  hipMemsetAsync(m1,   0, (size_t)N * H * 4, stream);
  hipMemsetAsync(dn1,  0, (size_t)N * H * 4, stream);
  hipMemsetAsync(out,  0, (size_t)N * D * 4, stream);
  hipMemsetAsync(m2,   0, (size_t)N * 4, stream);
  hipMemsetAsync(dn2,  0, (size_t)N * 4, stream);

  // =================== Layer 1 (H=8, C=64) ===================
  {
    int ntiles = (N / 16) * (F1 / 16);
    gemm_wmma_f32<64, 512><<<(ntiles + 7) / 8, T, 0, stream>>>(x, W1, xW1, N);
  }
  node_alpha<<<(N * H * 32 + T - 1) / T, T, 0, stream>>>(xW1, asrc1, adst1, as1, ad1, N, H);
  edge_leaky_max<8><<<(E * H + T - 1) / T, T, 0, stream>>>(as1, ad1, ei, e1, (unsigned*)m1, E);
  decode_max<<<(N * H + T - 1) / T, T, 0, stream>>>((unsigned*)m1, N * H);
  edge_expsum<8><<<(E * H + T - 1) / T, T, 0, stream>>>(e1, m1, ei, dn1, E);
  edge_norm<8><<<(E * H + T - 1) / T, T, 0, stream>>>(e1, dn1, ei, E);
  edge_aggregate<8, 64><<<(E * 128 + T - 1) / T, T, 0, stream>>>(xW1, e1, ei, out1, E);
  bias_act<512, true><<<((size_t)N * F1 + T - 1) / T, T, 0, stream>>>(out1, b1, N * F1);

  // =================== Layer 2 (H=1, C=64) ===================
  {
    int ntiles = (N / 16) * (D / 16);
    gemm_wmma_f32<512, 64><<<(ntiles + 7) / 8, T, 0, stream>>>(out1, W2, hW2, N);
  }
  node_alpha<<<(N * 32 + T - 1) / T, T, 0, stream>>>(hW2, asrc2, adst2, as2, ad2, N, 1);
  edge_leaky_max<1><<<(E + T - 1) / T, T, 0, stream>>>(as2, ad2, ei, e2, (unsigned*)m2, E);
  decode_max<<<(N + T - 1) / T, T, 0, stream>>>((unsigned*)m2, N);
  edge_expsum<1><<<(E + T - 1) / T, T, 0, stream>>>(e2, m2, ei, dn2, E);
  edge_norm<1><<<(E + T - 1) / T, T, 0, stream>>>(e2, dn2, ei, E);
  edge_aggregate<1, 64><<<(E * 16 + T - 1) / T, T, 0, stream>>>(hW2, e2, ei, out, E);
  bias_act<64, false><<<((size_t)N * D + T - 1) / T, T, 0, stream>>>(out, b2, N * D);
}